// YoloLoss_2585570312627
// MI455X (gfx1250) — compile-verified
//
#include <hip/hip_runtime.h>
#include <hip/hip_bf16.h>

typedef float v2f __attribute__((ext_vector_type(2)));
typedef float v8f __attribute__((ext_vector_type(8)));

#define SGRID 7
#define CH 30
#define LAMBDA_COORD 5.0f
#define LAMBDA_NOOBJ 0.5f

__device__ __forceinline__ void convert_box(float x, float y, float w, float h,
                                            float gi, float gj,
                                            float& x1, float& y1, float& x2, float& y2) {
  const float STEP = 1.0f / 7.0f;
  float cx = (x + gi) * STEP;
  float cy = (y + gj) * STEP;
  x1 = fmaxf(cx - w * 0.5f, 0.0f);
  y1 = fmaxf(cy - h * 0.5f, 0.0f);
  x2 = fminf(cx + w * 0.5f, 1.0f);
  y2 = fminf(cy + h * 0.5f, 1.0f);
}

__device__ __forceinline__ float iou_f(float a, float b, float c, float d,
                                       float q, float w, float e, float r) {
  float minx = fmaxf(a, q);
  float miny = fmaxf(b, w);
  float maxx = fminf(c, e);
  float maxy = fminf(d, r);
  float inter = (maxy - miny) * (maxx - minx);   // faithful: no clamp
  float uni = (e - q) * (r - w) + (d - b) * (c - a) - inter;
  return inter > 0.0f ? inter / (uni + 1e-5f) : 0.0f;
}

// Pass 1: one cell per thread; per-cell loss; wave reduce via V_WMMA_F32_16X16X4_F32;
// per-block partial to d_ws.
__global__ void __launch_bounds__(256) yolo_loss_main(const float* __restrict__ pred,
                                                      const float* __restrict__ target,
                                                      float* __restrict__ partial,
                                                      int ncells) {
  int cell = blockIdx.x * blockDim.x + threadIdx.x;
  float acc = 0.0f;
  if (cell < ncells) {
    float p[CH], t[CH];
    // 120-byte cell stride is 8B-aligned -> widest legal vector load is b64.
    const float2* pp = reinterpret_cast<const float2*>(pred)   + (long long)cell * (CH / 2);
    const float2* tp = reinterpret_cast<const float2*>(target) + (long long)cell * (CH / 2);
#pragma unroll
    for (int k = 0; k < CH / 2; ++k) {
      float2 v = pp[k]; p[2 * k] = v.x; p[2 * k + 1] = v.y;
      float2 u = tp[k]; t[2 * k] = u.x; t[2 * k + 1] = u.y;
    }
    int rem = cell % (SGRID * SGRID);
    float gi = (float)(rem / SGRID);   // first spatial axis -> x offset
    float gj = (float)(rem % SGRID);   // second spatial axis -> y offset

    float tx1, ty1, tx2, ty2, ax1, ay1, ax2, ay2, bx1, by1, bx2, by2;
    convert_box(t[0], t[1], t[2], t[3], gi, gj, tx1, ty1, tx2, ty2);
    convert_box(p[0], p[1], p[2], p[3], gi, gj, ax1, ay1, ax2, ay2);
    convert_box(p[5], p[6], p[7], p[8], gi, gj, bx1, by1, bx2, by2);
    float iou1 = iou_f(tx1, ty1, tx2, ty2, ax1, ay1, ax2, ay2);
    float iou2 = iou_f(tx1, ty1, tx2, ty2, bx1, by1, bx2, by2);

    bool sel2 = (iou1 <= iou2);
    float conf_t = sel2 ? iou2 : iou1;
    float px = sel2 ? p[5] : p[0];
    float py = sel2 ? p[6] : p[1];
    float pw = sel2 ? p[7] : p[2];
    float ph = sel2 ? p[8] : p[3];
    float pconf = sel2 ? p[9] : p[4];

    float obj   = (t[4] > 0.0f)  ? 1.0f : 0.0f;
    float noobj = (t[4] == 0.0f) ? 1.0f : 0.0f;

    float dc = pconf - conf_t;
    float obj_loss = dc * dc;
    float dx = px - t[0], dy = py - t[1];
    float dw = pw - t[2], dh = ph - t[3];
    float coord = dx * dx + dy * dy + dw * dw + dh * dh;
    float cls = 0.0f;
#pragma unroll
    for (int k = 10; k < CH; ++k) {
      float dd = p[k] - t[k];
      cls = fmaf(dd, dd, cls);
    }
    float d4 = p[4] - t[4], d9 = p[9] - t[9];
    float no = d4 * d4 + d9 * d9;

    acc = obj * (obj_loss + LAMBDA_COORD * coord + cls) + noobj * (LAMBDA_NOOBJ * no);
  }

  // ---- wave32 reduction on the matrix unit: D = A(16x4) * ones(4x16) ----
  // A layout (32-bit A, 16x4): lanes 0-15 VGPR0=K0, VGPR1=K1; lanes 16-31 VGPR0=K2, VGPR1=K3.
  // Put acc in VGPR0 slot, 0 in VGPR1 -> D[m][n] = acc[m] + acc[m+16].
  v2f a;     a[0] = acc;  a[1] = 0.0f;
  v2f bones; bones[0] = 1.0f; bones[1] = 1.0f;  // all-ones B: layout-independent
  v8f cz = {};
  v8f d = __builtin_amdgcn_wmma_f32_16x16x4_f32(
      /*neg_a=*/false, a, /*neg_b=*/false, bones,
      /*c_mod=*/(short)0, cz, /*reuse_a=*/false, /*reuse_b=*/false);
  // Lanes 0-15 hold rows 0-7 of D in VGPRs 0-7, lanes 16-31 hold rows 8-15.
  float part = d[0] + d[1] + d[2] + d[3] + d[4] + d[5] + d[6] + d[7];
  part += __shfl_xor(part, 16, 32);   // combine the two half-wave row groups

  __shared__ float lds[8];
  unsigned lane = threadIdx.x & 31u;
  unsigned wid  = threadIdx.x >> 5;
  if (lane == 0) lds[wid] = part;
  __syncthreads();
  if (threadIdx.x == 0) {
    float s = 0.0f;
#pragma unroll
    for (int i = 0; i < 8; ++i) s += lds[i];
    partial[blockIdx.x] = s;
  }
}

// Pass 2: deterministic final reduction in fp64, scale by 1/B.
__global__ void __launch_bounds__(256) yolo_loss_finish(const float* __restrict__ partial,
                                                        int n, float* __restrict__ out,
                                                        double invB) {
  __shared__ double lds[256];
  double s = 0.0;
  for (int i = threadIdx.x; i < n; i += 256) s += (double)partial[i];
  lds[threadIdx.x] = s;
  __syncthreads();
#pragma unroll
  for (int off = 128; off > 0; off >>= 1) {
    if ((int)threadIdx.x < off) lds[threadIdx.x] += lds[threadIdx.x + off];
    __syncthreads();
  }
  if (threadIdx.x == 0) out[0] = (float)(lds[0] * invB);
}

extern "C" void kernel_launch(void* const* d_in, const int* in_sizes, int n_in,
                              void* d_out, int out_size, void* d_ws, size_t ws_size,
                              hipStream_t stream) {
  const float* pred   = (const float*)d_in[0];
  const float* target = (const float*)d_in[1];
  int ncells = in_sizes[0] / CH;              // B * S * S
  int Bbatch = ncells / (SGRID * SGRID);      // 16384
  float* partial = (float*)d_ws;

  int nb = (ncells + 255) / 256;              // 3136 blocks
  yolo_loss_main<<<nb, 256, 0, stream>>>(pred, target, partial, ncells);
  yolo_loss_finish<<<1, 256, 0, stream>>>(partial, nb, (float*)d_out,
                                          1.0 / (double)Bbatch);
}